// BDRRAA_56453050139080
// MI455X (gfx1250) — compile-verified
//
#include <hip/hip_runtime.h>
#include <math.h>

#define N_I 50000
#define N_J 50000
#define S_I 3000
#define S_J 3000
#define SPAD 3008          // 3000 padded to 188*16
#define TILES 188
#define N_EDGES 500000
#define EPS 1e-6f

typedef float v2f __attribute__((ext_vector_type(2)));
typedef float v8f __attribute__((ext_vector_type(8)));

__device__ __forceinline__ float fast_exp(float x) { return __expf(x); }          // v_exp_f32
__device__ __forceinline__ float fast_sqrt(float x) { return __builtin_amdgcn_sqrtf(x); } // v_sqrt_f32

__device__ __forceinline__ void softmax8(float* z) {
    float mx = z[0];
#pragma unroll
    for (int k = 1; k < 8; ++k) mx = fmaxf(mx, z[k]);
    float s = 0.f;
#pragma unroll
    for (int k = 0; k < 8; ++k) { z[k] = fast_exp(z[k] - mx); s += z[k]; }
    float inv = 1.f / s;
#pragma unroll
    for (int k = 0; k < 8; ++k) z[k] *= inv;
}

// ---------------- init: zero f64 accumulators ----------------
// ws doubles: [0..1]=acc(mat, links), [2..9]=colsum, [10..73]=M8
__global__ void initKernel(double* ws) {
    int t = threadIdx.x;
    if (t < 74) ws[t] = 0.0;
}

// ---------------- pass A: softmax, sigmoid, colsum + Z (x) ZG outer product ----------------
__global__ void passAKernel(const float* __restrict__ Zi, const float* __restrict__ Zj,
                            const float* __restrict__ G, double* __restrict__ colsum,
                            double* __restrict__ M8) {
    __shared__ double sc[8];
    __shared__ double sm[64];
    int t = threadIdx.x;
    if (t < 8) sc[t] = 0.0;
    if (t < 64) sm[t] = 0.0;
    __syncthreads();
    int n = blockIdx.x * blockDim.x + t;
    if (n < N_I + N_J) {
        float z[8];
        if (n < N_I) {
#pragma unroll
            for (int k = 0; k < 8; ++k) z[k] = Zi[k * N_I + n];
        } else {
            int m = n - N_I;
#pragma unroll
            for (int k = 0; k < 8; ++k) z[k] = Zj[k * N_J + m];
        }
        softmax8(z);
        float zg[8];
#pragma unroll
        for (int k = 0; k < 8; ++k) {
            float g = 1.f / (1.f + fast_exp(-G[n * 8 + k]));
            zg[k] = z[k] * g;
            atomicAdd(&sc[k], (double)zg[k]);
        }
#pragma unroll
        for (int a = 0; a < 8; ++a)
#pragma unroll
            for (int b = 0; b < 8; ++b)
                atomicAdd(&sm[a * 8 + b], (double)(z[a] * zg[b]));
    }
    __syncthreads();
    if (t < 8) atomicAdd(&colsum[t], sc[t]);
    if (t < 64) atomicAdd(&M8[t], sm[t]);
}

// ---------------- AZC[r][c] = sum_m A[c][m] * M8[m][r] / colsum[r] ----------------
__global__ void azcKernel(const float* __restrict__ A, const double* __restrict__ colsum,
                          const double* __restrict__ M8, float* __restrict__ AZC) {
    int t = threadIdx.x;
    if (t < 64) {
        int r = t >> 3, c = t & 7;
        double inv = 1.0 / colsum[r];
        double acc = 0.0;
#pragma unroll
        for (int m = 0; m < 8; ++m) acc += (double)A[c * 8 + m] * (M8[m * 8 + r] * inv);
        AZC[t] = (float)acc;
    }
}

// ---------------- sampled latent positions + norms ----------------
// isI=1: PQ[s] = sum_r (m_r + eps)^2 ; isI=0: PQ[s] = sum m_r^2 - 2*eps*sum m_r
__global__ void sampleKernel(const float* __restrict__ Z, int ncols, const int* __restrict__ idx,
                             const float* __restrict__ bvec, const float* __restrict__ AZC,
                             float* __restrict__ M, float* __restrict__ PQ,
                             float* __restrict__ Bout, int S, int isI) {
    int s = blockIdx.x * blockDim.x + threadIdx.x;
    if (s >= SPAD) return;
    if (s >= S) {
#pragma unroll
        for (int r = 0; r < 8; ++r) M[s * 8 + r] = 0.f;
        PQ[s] = 0.f;
        Bout[s] = -INFINITY;     // exp(-inf - dist) == 0 for padded rows/cols
        return;
    }
    int c = idx[s];
    float z[8];
#pragma unroll
    for (int k = 0; k < 8; ++k) z[k] = Z[k * ncols + c];
    softmax8(z);
    float a1 = 0.f, a2 = 0.f;
#pragma unroll
    for (int r = 0; r < 8; ++r) {
        float m = 0.f;
#pragma unroll
        for (int k = 0; k < 8; ++k) m = fmaf(AZC[r * 8 + k], z[k], m);
        M[s * 8 + r] = m;
        if (isI) { float tt = m + EPS; a1 = fmaf(tt, tt, a1); }
        else     { a1 = fmaf(m, m, a1); a2 += m; }
    }
    PQ[s] = isI ? a1 : (a1 - 2.f * EPS * a2);
    Bout[s] = bvec[c];
}

// ---------------- pairwise tile kernel: WMMA f32 16x16x4 (rank-8 Gram) ----------------
// dist^2(i,j) = P(i) + Q(j) - 2 * (Mi @ Mj^T)(i,j)   (exact expansion of ((x-y)+eps)^2)
__global__ void pairKernel(const float* __restrict__ Mi, const float* __restrict__ Pv,
                           const float* __restrict__ Bv,
                           const float* __restrict__ Mj, const float* __restrict__ Qv,
                           const float* __restrict__ Gv, double* __restrict__ acc) {
    __shared__ double bsum;
    if (threadIdx.x == 0) bsum = 0.0;
    __syncthreads();

    const int lane = threadIdx.x & 31;
    const int wave = threadIdx.x >> 5;
    const int tile = blockIdx.x * 8 + wave;          // grid covers tiles exactly
    const int ti = tile / TILES;
    const int tj = tile - ti * TILES;
    const int half = lane >> 4;                      // 0: lanes 0-15, 1: lanes 16-31
    const int r = lane & 15;

    // A frag (16x4 f32, MxK): lanes 0-15 -> K={0,1}, lanes 16-31 -> K={2,3}, M=lane%16
    // B frag (4x16 f32, KxN): transpose-symmetric -> same load pattern from row-major Mj
    const float* ar = Mi + (size_t)(ti * 16 + r) * 8 + half * 2;
    const float* br = Mj + (size_t)(tj * 16 + r) * 8 + half * 2;
    v2f a0 = { ar[0], ar[1] };
    v2f b0 = { br[0], br[1] };
    v2f a1 = { ar[4], ar[5] };
    v2f b1 = { br[4], br[5] };

    v8f c = {};
    c = __builtin_amdgcn_wmma_f32_16x16x4_f32(false, a0, false, b0, (short)0, c, false, false);
    c = __builtin_amdgcn_wmma_f32_16x16x4_f32(false, a1, false, b1, (short)0, c, false, false);

    // D layout: VGPR v -> M = v + 8*half, N = lane%16
    const int nn = lane & 15;
    const float q   = Qv[tj * 16 + nn];
    const float gma = Gv[tj * 16 + nn];
    double local = 0.0;
#pragma unroll
    for (int v = 0; v < 8; ++v) {
        const int m = v + 8 * half;
        const float p = Pv[ti * 16 + m];
        const float bias = Bv[ti * 16 + m] + gma;
        const float d2 = fmaxf(fmaf(-2.0f, c[v], p + q), 0.0f);
        local += (double)fast_exp(bias - fast_sqrt(d2));
    }
    atomicAdd(&bsum, local);
    __syncthreads();
    if (threadIdx.x == 0) atomicAdd(&acc[0], bsum);
}

// ---------------- sparse edge term ----------------
__global__ void edgeKernel(const float* __restrict__ Zi, const float* __restrict__ Zj,
                           const float* __restrict__ beta, const float* __restrict__ gamma,
                           const int* __restrict__ ei, const int* __restrict__ ej,
                           const float* __restrict__ AZC, double* __restrict__ acc) {
    __shared__ double bsum;
    if (threadIdx.x == 0) bsum = 0.0;
    __syncthreads();
    int e = blockIdx.x * blockDim.x + threadIdx.x;
    double local = 0.0;
    if (e < N_EDGES) {
        int ci = ei[e], cj = ej[e];
        float zi[8], zj[8];
#pragma unroll
        for (int k = 0; k < 8; ++k) { zi[k] = Zi[k * N_I + ci]; zj[k] = Zj[k * N_J + cj]; }
        softmax8(zi);
        softmax8(zj);
        float sq = 0.f;
#pragma unroll
        for (int r = 0; r < 8; ++r) {
            float mi = 0.f, mj = 0.f;
#pragma unroll
            for (int k = 0; k < 8; ++k) {
                mi = fmaf(AZC[r * 8 + k], zi[k], mi);
                mj = fmaf(AZC[r * 8 + k], zj[k], mj);
            }
            float d = (mi - mj) + EPS;
            sq = fmaf(d, d, sq);
        }
        local = (double)beta[ci] + (double)gamma[cj] - (double)sq;
    }
    atomicAdd(&bsum, local);
    __syncthreads();
    if (threadIdx.x == 0) atomicAdd(&acc[1], bsum);
}

// ---------------- final scalar ----------------
__global__ void finalKernel(const double* __restrict__ acc, float* __restrict__ out) {
    out[0] = (float)(acc[1] - acc[0]);   // mat_links - mat
}

extern "C" void kernel_launch(void* const* d_in, const int* in_sizes, int n_in,
                              void* d_out, int out_size, void* d_ws, size_t ws_size,
                              hipStream_t stream) {
    (void)in_sizes; (void)n_in; (void)out_size; (void)ws_size;
    const float* beta  = (const float*)d_in[0];
    const float* gamma = (const float*)d_in[1];
    const float* A     = (const float*)d_in[2];
    const float* Z_i   = (const float*)d_in[3];
    const float* Z_j   = (const float*)d_in[4];
    const float* G     = (const float*)d_in[5];
    const int*   si    = (const int*)d_in[6];
    const int*   sj    = (const int*)d_in[7];
    const int*   ei    = (const int*)d_in[8];
    const int*   ej    = (const int*)d_in[9];

    char* ws = (char*)d_ws;
    double* acc    = (double*)ws;          // [2]
    double* colsum = acc + 2;              // [8]
    double* M8     = colsum + 8;           // [64]
    float*  AZC    = (float*)(ws + 1024);  // [64]
    float*  Mi     = (float*)(ws + 2048);  // [SPAD*8]
    float*  Pv     = Mi + SPAD * 8;        // [SPAD]
    float*  Bv     = Pv + SPAD;            // [SPAD]
    float*  Mj     = Bv + SPAD;            // [SPAD*8]
    float*  Qv     = Mj + SPAD * 8;        // [SPAD]
    float*  Gv     = Qv + SPAD;            // [SPAD]

    initKernel<<<1, 128, 0, stream>>>(acc);
    passAKernel<<<(N_I + N_J + 255) / 256, 256, 0, stream>>>(Z_i, Z_j, G, colsum, M8);
    azcKernel<<<1, 64, 0, stream>>>(A, colsum, M8, AZC);
    sampleKernel<<<(SPAD + 127) / 128, 128, 0, stream>>>(Z_i, N_I, si, beta,  AZC, Mi, Pv, Bv, S_I, 1);
    sampleKernel<<<(SPAD + 127) / 128, 128, 0, stream>>>(Z_j, N_J, sj, gamma, AZC, Mj, Qv, Gv, S_J, 0);
    pairKernel<<<(TILES * TILES) / 8, 256, 0, stream>>>(Mi, Pv, Bv, Mj, Qv, Gv, acc);
    edgeKernel<<<(N_EDGES + 255) / 256, 256, 0, stream>>>(Z_i, Z_j, beta, gamma, ei, ej, AZC, acc);
    finalKernel<<<1, 1, 0, stream>>>(acc, (float*)d_out);
}